// PDFSampler_30468497998318
// MI455X (gfx1250) — compile-verified
//
#include <hip/hip_runtime.h>
#include <stdint.h>

// Problem constants (must match reference)
#define RAYS   262144
#define NW     64          // histogram bins per ray
#define NB     65          // cdf entries == edge count == sample count
#define NOUT   130         // merged output per ray
#define WPB    8           // waves (rays) per 256-thread block
#define HPAD   0.01f
#define EPSV   1e-5f

typedef __attribute__((ext_vector_type(2))) float vfloat2;

__global__ __launch_bounds__(256) void PDFSampler_kernel(
    const float* __restrict__ weights,   // (R, 64)
    const float* __restrict__ starts,    // (R, 64)  = edges[:, :-1]
    const float* __restrict__ ends,      // (R, 64)  = edges[:, 1:]
    const float* __restrict__ nears,     // (R)
    const float* __restrict__ fars,      // (R)
    float* __restrict__ out)             // (2, R, 130)
{
    __shared__ float s_cdf[WPB][NB + 1];                 // 65 used
    __shared__ __align__(16) float s_eb[WPB][72];        // 65 used, 8B-aligned per lane pair
    __shared__ float s_smp[WPB][NB + 1];                 // 65 used
    __shared__ __align__(16) float s_out[WPB][NOUT + 2]; // 130 used, slot stride 528B (8B-mult)

    const int lane = threadIdx.x & 31;
    const int slot = threadIdx.x >> 5;
    const int ray  = blockIdx.x * WPB + slot;

    const float* wptr = weights + (size_t)ray * NW;
    const float* sptr = starts  + (size_t)ray * NW;

    // ---- Stage the 64 sorted bin-starts straight into LDS via the CDNA5
    //      async global->LDS path (ASYNCcnt). One b64 per lane covers all 64.
    {
        unsigned lds_a = (unsigned)(uintptr_t)&s_eb[slot][2 * lane];
        unsigned long long g_a = (unsigned long long)(uintptr_t)(sptr + 2 * lane);
        asm volatile("global_load_async_to_lds_b64 %0, %1, off"
                     :: "v"(lds_a), "v"(g_a) : "memory");
    }
    if (lane == 0) s_eb[slot][64] = ends[(size_t)ray * NW + (NW - 1)];

    // ---- weights -> pdf (overlaps with async edge staging); NT: read-once ----
    vfloat2 w2 = __builtin_nontemporal_load((const vfloat2*)wptr + lane);
    float w0 = w2.x + HPAD;
    float w1 = w2.y + HPAD;

    float tot = w0 + w1;                       // wave32 reduction for w_sum
    #pragma unroll
    for (int m = 16; m >= 1; m >>= 1) tot += __shfl_xor(tot, m, 32);

    float padv = fmaxf(EPSV - tot, 0.0f);      // relu(EPS - w_sum), 0 in practice
    w0 += padv * (1.0f / NW);
    w1 += padv * (1.0f / NW);
    tot += padv;
    float p0 = w0 / tot;
    float p1 = w1 / tot;

    // inclusive wave32 scan of lane-pair sums -> cdf
    float ps = p0 + p1;
    float sc = ps;
    #pragma unroll
    for (int d = 1; d < 32; d <<= 1) {
        float t = __shfl_up(sc, d, 32);
        if (lane >= d) sc += t;
    }
    float excl = sc - ps;
    float c0 = fminf(excl + p0, 1.0f);
    float c1 = fminf(sc, 1.0f);
    if (lane == 0) s_cdf[slot][0] = 0.0f;
    s_cdf[slot][2 * lane + 1] = c0;
    s_cdf[slot][2 * lane + 2] = c1;

    asm volatile("s_wait_asynccnt 0" ::: "memory");   // edges now resident in LDS
    __syncthreads();

    const float* cdf = s_cdf[slot];
    const float* eb  = s_eb[slot];

    // ---- inverse-CDF sampling: 65 stratified u values ----
    for (int j = lane; j < NB; j += 32) {
        float u = ((float)j + 0.5f) * (1.0f / 65.0f);     // linspace + half-step
        int lo = 0, hi = NB;                               // searchsorted side='right'
        while (lo < hi) { int mid = (lo + hi) >> 1; if (cdf[mid] <= u) lo = mid + 1; else hi = mid; }
        int below = lo - 1; below = below < 0 ? 0 : (below > NB - 1 ? NB - 1 : below);
        int above = lo > NB - 1 ? NB - 1 : lo;
        float cg0 = cdf[below], cg1 = cdf[above];
        float bg0 = eb[below],  bg1 = eb[above];
        float den = cg1 - cg0;
        // nan_to_num(nan->0, inf->huge) then clip(0,1):
        float t = (den > 0.0f) ? (u - cg0) / den : ((u > cg0) ? 1.0f : 0.0f);
        t = fminf(fmaxf(t, 0.0f), 1.0f);
        s_smp[slot][j] = bg0 + t * (bg1 - bg0);
    }
    __syncthreads();

    const float* sm = s_smp[slot];

    // ---- merge two sorted 65-lists (== reference's sort of concat) via ranks ----
    for (int i = lane; i < NB; i += 32) {
        // edge eb[i] goes to i + #(samples < eb[i])
        float a = eb[i];
        int lo = 0, hi = NB;
        while (lo < hi) { int mid = (lo + hi) >> 1; if (sm[mid] < a) lo = mid + 1; else hi = mid; }
        s_out[slot][i + lo] = a;
        // sample sm[i] goes to i + #(edges <= sm[i])   (ties: edge first -> unique slots)
        float b = sm[i];
        lo = 0; hi = NB;
        while (lo < hi) { int mid = (lo + hi) >> 1; if (eb[mid] <= b) lo = mid + 1; else hi = mid; }
        s_out[slot][i + lo] = b;
    }
    __syncthreads();

    // ---- coalesced b64 write-out with NT hint (streaming, never re-read) ----
    const float nr = nears[ray];
    const float fr = fars[ray];
    const float dfn = fr - nr;
    // row byte offset = ray*520 -> always 8B aligned
    vfloat2* ob2 = (vfloat2*)(out + (size_t)ray * NOUT);
    vfloat2* oe2 = (vfloat2*)(out + (size_t)RAYS * NOUT + (size_t)ray * NOUT);
    const vfloat2* so2 = (const vfloat2*)s_out[slot];
    for (int k = lane; k < NOUT / 2; k += 32) {           // 65 float2 per ray
        vfloat2 b2 = so2[k];
        __builtin_nontemporal_store(b2, ob2 + k);
        vfloat2 e2;
        e2.x = fmaf(b2.x, dfn, nr);
        e2.y = fmaf(b2.y, dfn, nr);
        __builtin_nontemporal_store(e2, oe2 + k);
    }
}

extern "C" void kernel_launch(void* const* d_in, const int* in_sizes, int n_in,
                              void* d_out, int out_size, void* d_ws, size_t ws_size,
                              hipStream_t stream) {
    (void)in_sizes; (void)n_in; (void)out_size; (void)d_ws; (void)ws_size;
    const float* weights = (const float*)d_in[0];
    const float* starts  = (const float*)d_in[1];
    const float* ends    = (const float*)d_in[2];
    const float* nears   = (const float*)d_in[3];
    const float* fars    = (const float*)d_in[4];
    float* out = (float*)d_out;

    dim3 grid(RAYS / WPB);   // 32768 blocks, exact
    dim3 block(256);         // 8 wave32 = 8 rays per block
    hipLaunchKernelGGL(PDFSampler_kernel, grid, block, 0, stream,
                       weights, starts, ends, nears, fars, out);
}